// HierarchicalPooling_48172353192358
// MI455X (gfx1250) — compile-verified
//
#include <hip/hip_runtime.h>
#include <hip/hip_bf16.h>

// Problem constants (fixed by the reference)
#define HDIM 256
#define BDIM 8
#define FDIM 256
#define SDIM (FDIM * BDIM)   // 2048 flat segments, seg = fid*B + batch

typedef float v2f __attribute__((ext_vector_type(2)));
typedef float v8f __attribute__((ext_vector_type(8)));

// ---------------------------------------------------------------- K1: zero counts
__global__ void k_zero_counts(int* counts) {
    int i = blockIdx.x * blockDim.x + threadIdx.x;
    if (i < SDIM) counts[i] = 0;
}

// ---------------------------------------------------------------- K2: histogram (LDS privatized)
__global__ void k_hist(const int* __restrict__ batch, const int* __restrict__ fid,
                       int* __restrict__ counts, int N, int chunk) {
    __shared__ int lc[SDIM];
    for (int s = threadIdx.x; s < SDIM; s += blockDim.x) lc[s] = 0;
    __syncthreads();
    int begin = blockIdx.x * chunk;
    int end   = min(N, begin + chunk);
    for (int i = begin + (int)threadIdx.x; i < end; i += blockDim.x) {
        int s = fid[i] * BDIM + batch[i];
        atomicAdd(&lc[s], 1);
    }
    __syncthreads();
    for (int s = threadIdx.x; s < SDIM; s += blockDim.x)
        if (lc[s]) atomicAdd(&counts[s], lc[s]);
}

// ---------------------------------------------------------------- K3: exclusive scan (1 block)
__global__ void k_scan(const int* __restrict__ counts, int* __restrict__ offsets,
                       int* __restrict__ cursor) {
    __shared__ int sums[256];
    int tid  = threadIdx.x;
    int base = tid * 8;
    int loc[8];
    int tot = 0;
#pragma unroll
    for (int j = 0; j < 8; ++j) { loc[j] = counts[base + j]; tot += loc[j]; }
    sums[tid] = tot;
    __syncthreads();
    if (tid == 0) {
        int run = 0;
        for (int t = 0; t < 256; ++t) { int v = sums[t]; sums[t] = run; run += v; }
    }
    __syncthreads();
    int run = sums[tid];
#pragma unroll
    for (int j = 0; j < 8; ++j) {
        offsets[base + j] = run;
        cursor[base + j]  = run;
        run += loc[j];
    }
}

// ---------------------------------------------------------------- K4: scatter node ids (LDS privatized)
__global__ void k_scatter(const int* __restrict__ batch, const int* __restrict__ fid,
                          int* __restrict__ cursor, int* __restrict__ nodeidx,
                          int N, int chunk) {
    __shared__ int lc[SDIM];
    __shared__ int lbase[SDIM];
    for (int s = threadIdx.x; s < SDIM; s += blockDim.x) lc[s] = 0;
    __syncthreads();
    int begin = blockIdx.x * chunk;
    int end   = min(N, begin + chunk);
    // pass 1: local counts
    for (int i = begin + (int)threadIdx.x; i < end; i += blockDim.x) {
        int s = fid[i] * BDIM + batch[i];
        atomicAdd(&lc[s], 1);
    }
    __syncthreads();
    // reserve global ranges, reset local cursors
    for (int s = threadIdx.x; s < SDIM; s += blockDim.x) {
        int c = lc[s];
        lbase[s] = c ? atomicAdd(&cursor[s], c) : 0;
        lc[s] = 0;
    }
    __syncthreads();
    // pass 2: place
    for (int i = begin + (int)threadIdx.x; i < end; i += blockDim.x) {
        int s = fid[i] * BDIM + batch[i];
        int p = atomicAdd(&lc[s], 1);
        nodeidx[lbase[s] + p] = i;
    }
}

// ---------------------------------------------------------------- K5: gather-reduce (bandwidth pass)
// one block per segment; thread = column h; x read exactly once, coalesced
__global__ void k_gather(const float* __restrict__ x,
                         const int* __restrict__ offsets, const int* __restrict__ counts,
                         const int* __restrict__ nodeidx, float* __restrict__ feat) {
    __shared__ int sidx[64];
    int s   = blockIdx.x;
    int h   = threadIdx.x;
    int beg = offsets[s];
    int cnt = counts[s];
    float sum = 0.0f;
    float mx  = -__builtin_inff();
    for (int base = 0; base < cnt; base += 64) {
        int m = min(64, cnt - base);
        if ((int)threadIdx.x < m) sidx[threadIdx.x] = nodeidx[beg + base + threadIdx.x];
        __syncthreads();
        for (int j = 0; j < m; ++j) {
            float v = x[(size_t)sidx[j] * HDIM + h];
            sum += v;
            mx = fmaxf(mx, v);
        }
        __syncthreads();
    }
    float inv = 1.0f / (float)max(cnt, 1);
    feat[(size_t)s * (2 * HDIM) + h]        = sum * inv;
    feat[(size_t)s * (2 * HDIM) + HDIM + h] = (cnt > 0) ? mx : 0.0f;
}

// ---------------------------------------------------------------- K6: GEMM1 via V_WMMA_F32_16X16X4_F32
// func[m,h] = relu( sum_k feat[m,k] * Wf[h,k] + bf[h] ),  M=2048, K=512, Nout=256
__global__ void k_gemm_wmma(const float* __restrict__ feat, const float* __restrict__ Wf,
                            const float* __restrict__ bf, float* __restrict__ func) {
    const int wave  = blockIdx.x * (blockDim.x / 32) + (threadIdx.x >> 5);
    const int lane  = threadIdx.x & 31;
    const int half  = lane >> 4;       // 0: lanes 0-15, 1: lanes 16-31
    const int l16   = lane & 15;
    const int tileM = wave >> 4;       // 128 M-tiles
    const int tileN = wave & 15;       // 16  N-tiles

    const int m0 = tileM * 16 + l16;   // A row owned by this lane
    const int n0 = tileN * 16 + l16;   // B column (Wf row) owned by this lane

    const float* Arow = feat + (size_t)m0 * (2 * HDIM);
    const float* Brow = Wf   + (size_t)n0 * (2 * HDIM);   // Wf is (H, 2H) row-major

    v8f c = {0.f, 0.f, 0.f, 0.f, 0.f, 0.f, 0.f, 0.f};
#pragma unroll 4
    for (int k = 0; k < 2 * HDIM; k += 4) {
        int kk = k + half * 2;                 // lanes 0-15: K=k,k+1; lanes 16-31: K=k+2,k+3
        v2f a, b;
        a.x = Arow[kk]; a.y = Arow[kk + 1];
        b.x = Brow[kk]; b.y = Brow[kk + 1];
        c = __builtin_amdgcn_wmma_f32_16x16x4_f32(
                /*neg_a=*/false, a, /*neg_b=*/false, b,
                /*c_mod=*/(short)0, c, /*reuse_a=*/false, /*reuse_b=*/false);
    }
    float bias = bf[tileN * 16 + l16];
#pragma unroll
    for (int i = 0; i < 8; ++i) {
        int row = tileM * 16 + i + half * 8;   // C layout: VGPR i -> M = i (+8 for upper lanes)
        float v = fmaxf(c[i] + bias, 0.0f);
        func[(size_t)row * HDIM + tileN * 16 + l16] = v;
    }
}

// ---------------------------------------------------------------- K7: mean+max over functions
__global__ void k_filefeat(const float* __restrict__ func, float* __restrict__ filefeat) {
    int b = blockIdx.x;      // 0..7
    int h = threadIdx.x;     // 0..255
    float sum = 0.0f;
    float mx  = -__builtin_inff();
    for (int f = 0; f < FDIM; ++f) {
        float v = func[((size_t)f * BDIM + b) * HDIM + h];
        sum += v;
        mx = fmaxf(mx, v);
    }
    filefeat[b * (2 * HDIM) + h]        = sum * (1.0f / (float)FDIM);
    filefeat[b * (2 * HDIM) + HDIM + h] = mx;
}

// ---------------------------------------------------------------- K8: final tiny GEMM (1 block)
__global__ void k_final(const float* __restrict__ filefeat, const float* __restrict__ Wg,
                        const float* __restrict__ bg, float* __restrict__ out) {
    __shared__ float ff[BDIM * 2 * HDIM];      // 16 KB
    for (int i = threadIdx.x; i < BDIM * 2 * HDIM; i += blockDim.x) ff[i] = filefeat[i];
    __syncthreads();
    int h = threadIdx.x;
    float acc[BDIM];
#pragma unroll
    for (int b = 0; b < BDIM; ++b) acc[b] = bg[h];
    const float* w = Wg + (size_t)h * (2 * HDIM);
    for (int k = 0; k < 2 * HDIM; ++k) {
        float wv = w[k];
#pragma unroll
        for (int b = 0; b < BDIM; ++b) acc[b] += ff[b * (2 * HDIM) + k] * wv;
    }
#pragma unroll
    for (int b = 0; b < BDIM; ++b) out[b * HDIM + h] = fmaxf(acc[b], 0.0f);
}

// ----------------------------------------------------------------
extern "C" void kernel_launch(void* const* d_in, const int* in_sizes, int n_in,
                              void* d_out, int out_size, void* d_ws, size_t ws_size,
                              hipStream_t stream) {
    const float* x     = (const float*)d_in[0];
    const int*   batch = (const int*)  d_in[1];
    const int*   fid   = (const int*)  d_in[2];
    const float* Wf    = (const float*)d_in[3];
    const float* bf    = (const float*)d_in[4];
    const float* Wg    = (const float*)d_in[5];
    const float* bg    = (const float*)d_in[6];
    float* out = (float*)d_out;

    const int N = in_sizes[1];   // number of nodes

    // workspace layout (all 4-byte aligned)
    int*   counts   = (int*)d_ws;
    int*   offsets  = counts  + SDIM;
    int*   cursor   = offsets + SDIM;
    int*   nodeidx  = cursor  + SDIM;
    float* feat     = (float*)(nodeidx + N);              // (S, 2H)  4 MB
    float* func     = feat + (size_t)SDIM * 2 * HDIM;     // (S, H)   2 MB
    float* filefeat = func + (size_t)SDIM * HDIM;         // (B, 2H)  16 KB

    const int nb    = 64;
    const int chunk = (N + nb - 1) / nb;

    k_zero_counts<<<(SDIM + 255) / 256, 256, 0, stream>>>(counts);
    k_hist      <<<nb, 256, 0, stream>>>(batch, fid, counts, N, chunk);
    k_scan      <<<1, 256, 0, stream>>>(counts, offsets, cursor);
    k_scatter   <<<nb, 256, 0, stream>>>(batch, fid, cursor, nodeidx, N, chunk);
    k_gather    <<<SDIM, 256, 0, stream>>>(x, offsets, counts, nodeidx, feat);
    k_gemm_wmma <<<256, 256, 0, stream>>>(feat, Wf, bf, func);   // 2048 waves = 2048 tiles
    k_filefeat  <<<BDIM, 256, 0, stream>>>(func, filefeat);
    k_final     <<<1, 256, 0, stream>>>(filefeat, Wg, bg, out);
}